// ChannelDistributionAlignmentLoss_84413287235931
// MI455X (gfx1250) — compile-verified
//
#include <hip/hip_runtime.h>
#include <math.h>

// ---------------------------------------------------------------------------
// ChannelDistributionAlignmentLoss for gfx1250 (MI455X).
// DFT-as-GEMM on the fp32 WMMA path (V_WMMA_F32_16X16X4_F32) + async
// global->LDS staging (GLOBAL_LOAD_ASYNC_TO_LDS_B128, ASYNCcnt).
// Memory-bound (~23 us HBM floor) so the matrix-unit DFT is essentially free
// and keeps full fp32 precision.
// ---------------------------------------------------------------------------

typedef __attribute__((ext_vector_type(2))) float v2f;
typedef __attribute__((ext_vector_type(8))) float v8f;

#define HDIM     128
#define WR       65          // rfft2 width = W/2 + 1
#define CCH      256
#define BTDIM    32
#define LSTRIDE  132         // padded LDS row stride (floats)
#define NB       8
#define USE_ASYNC_COPY 1

__device__ __forceinline__ v8f wmma_f32(v2f a, v2f b, v8f c) {
    // D = A(16x4) * B(4x16) + C(16x16), fp32 matrix core
    return __builtin_amdgcn_wmma_f32_16x16x4_f32(
        false, a, false, b, (short)0, c, false, false);
}

#if USE_ASYNC_COPY
// Per-lane 16B async copy global -> LDS (tracked by ASYNCcnt, no VGPR data).
__device__ __forceinline__ void async_copy_b128(unsigned lds_byte_off,
                                                const float4* gptr) {
    asm volatile("global_load_async_to_lds_b128 %0, %1, off"
                 :
                 : "v"(lds_byte_off), "v"(gptr)
                 : "memory");
}
__device__ __forceinline__ void async_wait0() {
    asm volatile("s_wait_asynccnt 0" ::: "memory");
}
#endif

// ---------------------------------------------------------------------------
// Kernel 1: one 128x128 image per workgroup.
//   band feats via 2D DFT (two complex GEMM stages) -> high_freq_ratio[n]
// ---------------------------------------------------------------------------
__global__ __launch_bounds__(128)
void spectral_band_kernel(const float* __restrict__ x, float* __restrict__ hfr)
{
    __shared__ __align__(16) float  A[HDIM * LSTRIDE];   // image tile (67.5 KB)
    __shared__ __align__(16) float  Brt[16 * LSTRIDE];   // row-DFT real
    __shared__ __align__(16) float  Bit[16 * LSTRIDE];   // row-DFT imag
    __shared__ __align__(8)  float2 tab2[128];           // (cos, sin) twiddles
    __shared__ float red[128][NB];

    const int tid  = threadIdx.x;
    const int lane = tid & 31;
    const int wave = tid >> 5;

    const int n  = blockIdx.x;        // image id in [0, 8192)
    const int bt = n >> 8;            // (b*T + t)
    const int c  = n & 255;
    const int bb = bt >> 2;           // b
    const int tt = bt & 3;            // t

    const float4* src = (const float4*)(x +
        ((size_t)(bb * CCH + c) * 65536u + (size_t)tt * 16384u));

    // stage image into LDS: per-lane b128 async copies, coalesced
#if USE_ASYNC_COPY
    {
        const unsigned abase = (unsigned)(unsigned long long)(void*)A;
        #pragma unroll
        for (int i = 0; i < 32; ++i) {
            int v = i * 128 + tid;            // float4 index, row-major
            int h = v >> 5;
            int w = (v & 31) << 2;
            async_copy_b128(abase + (unsigned)(h * LSTRIDE + w) * 4u, src + v);
        }
    }
#else
    #pragma unroll
    for (int i = 0; i < 32; ++i) {
        int v = i * 128 + tid;
        int h = v >> 5;
        int w = (v & 31) << 2;
        *(float4*)&A[h * LSTRIDE + w] = src[v];
    }
#endif

    // twiddle table: e^{-2*pi*i*p/128} = (tab2[p].x, -tab2[p].y)
    if (tid < 128) {
        float ang = (float)tid * 0.04908738521234052f;   // pi/64
        float s, cc;
        __sincosf(ang, &s, &cc);
        tab2[tid] = make_float2(cc, s);
    }

    float feats[NB];
    #pragma unroll
    for (int i = 0; i < NB; ++i) feats[i] = 0.0f;

#if USE_ASYNC_COPY
    async_wait0();
#endif
    __syncthreads();

    const int m     = lane & 15;          // M (or N) index within a fragment
    const int kq    = (lane >> 4) << 1;   // K sub-offset: 0 or 2
    const int rbase = (lane >> 4) << 3;
    const int c0    = wave * 32 + m;      // this wave's two column tiles
    const int c1    = c0 + 16;

    for (int kt = 0; kt < 8; ++kt) {
        const int k0 = kt * 16;

        // ---- Phase A: B(16x128) = E_H[k0..k0+15, :] * A  (re+im, 2 tiles) --
        {
            const int krow  = k0 + m;
            v8f aR0 = {}, aI0 = {}, aR1 = {}, aI1 = {};
            int p0          = (krow * kq) & 127;   // phase at (krow, h=kq)
            const int pinc  = krow & 127;          // h -> h+1
            const int pstep = (krow * 4) & 127;    // h -> h+4
            for (int kk = 0; kk < 32; ++kk) {
                const int h = kk * 4 + kq;
                float2 t0 = tab2[p0];
                float2 t1 = tab2[(p0 + pinc) & 127];
                v2f ec = { t0.x,  t1.x};           // Er = cos
                v2f es = {-t0.y, -t1.y};           // Ei = -sin
                v2f b0, b1;
                b0.x = A[h * LSTRIDE + c0];  b0.y = A[(h + 1) * LSTRIDE + c0];
                b1.x = A[h * LSTRIDE + c1];  b1.y = A[(h + 1) * LSTRIDE + c1];
                aR0 = wmma_f32(ec, b0, aR0);
                aI0 = wmma_f32(es, b0, aI0);
                aR1 = wmma_f32(ec, b1, aR1);
                aI1 = wmma_f32(es, b1, aI1);
                p0 = (p0 + pstep) & 127;
            }
            #pragma unroll
            for (int r = 0; r < 8; ++r) {
                const int row = (rbase + r) * LSTRIDE;
                Brt[row + c0] = aR0[r];  Bit[row + c0] = aI0[r];
                Brt[row + c1] = aR1[r];  Bit[row + c1] = aI1[r];
            }
        }
        __syncthreads();

        // ---- Phase B: F = B * E_W^T per l-tile, magnitude + band binning ---
        for (int lt = wave; lt < 5; lt += 4) {     // l = 0..79, bin l < 65
            const int l = lt * 16 + m;
            v8f aFr = {}, aFi = {};
            int p0          = (kq * l) & 127;      // phase at (w=kq, l)
            const int pinc  = l & 127;             // w -> w+1
            const int pstep = (l * 4) & 127;       // w -> w+4
            for (int kk = 0; kk < 32; ++kk) {
                const int wrow = kk * 4 + kq;
                float2 br2 = *(const float2*)&Brt[m * LSTRIDE + wrow];
                float2 bi2 = *(const float2*)&Bit[m * LSTRIDE + wrow];
                float2 t0  = tab2[p0];
                float2 t1  = tab2[(p0 + pinc) & 127];
                v2f br = {br2.x, br2.y};
                v2f bi = {bi2.x, bi2.y};
                v2f ct = { t0.x,  t1.x};           // cos
                v2f sp = { t0.y,  t1.y};           // +sin  (= -St)
                v2f sm = {-t0.y, -t1.y};           //  St
                aFr = wmma_f32(br, ct, aFr);       // Fr += Br*Ct
                aFr = wmma_f32(bi, sp, aFr);       // Fr -= Bi*St
                aFi = wmma_f32(br, sm, aFi);       // Fi += Br*St
                aFi = wmma_f32(bi, ct, aFi);       // Fi += Bi*Ct
                p0 = (p0 + pstep) & 127;
            }
            if (l < WR) {
                #pragma unroll
                for (int r = 0; r < 8; ++r) {
                    const int krow = k0 + rbase + r;
                    float fre = aFr[r], fim = aFi[r];
                    float mag = __fsqrt_rn(fre * fre + fim * fim + 1e-8f);
                    int jh = (krow + 64) & 127;          // fftshift (H)
                    int jw = (l + 32) % 65;              // fftshift (Wr)
                    float hg = (float)(jh - 64) * (1.0f / 128.0f);
                    float wg = (float)(jw - 32) * (1.0f / 65.0f);
                    float frad = __fsqrt_rn(hg * hg + wg * wg);
                    int band = (int)(frad * 8.0f);
                    band = band > 7 ? 7 : band;
                    #pragma unroll
                    for (int bnd = 0; bnd < NB; ++bnd)
                        feats[bnd] += (bnd == band) ? mag : 0.0f;
                }
            }
        }
        __syncthreads();
    }

    // block reduction of the 8 band sums
    #pragma unroll
    for (int bnd = 0; bnd < NB; ++bnd) red[tid][bnd] = feats[bnd];
    __syncthreads();
    for (int s = 64; s > 0; s >>= 1) {
        if (tid < s) {
            #pragma unroll
            for (int bnd = 0; bnd < NB; ++bnd)
                red[tid][bnd] += red[tid + s][bnd];
        }
        __syncthreads();
    }
    if (tid == 0) {
        float tot = 0.f, hi = 0.f;
        #pragma unroll
        for (int bnd = 0; bnd < NB; ++bnd) {
            tot += red[0][bnd];
            if (bnd >= NB / 2) hi += red[0][bnd];
        }
        hfr[n] = hi / (tot + 1e-8f);   // normalized high-freq ratio
    }
}

// ---------------------------------------------------------------------------
// Kernel 2: loss = 1 - || sum_bt f_bt ||^2 / (BT*BT),  f_bt = row-normalized
// (sum over all entries of f f^T equals |sum of rows|^2)
// ---------------------------------------------------------------------------
__global__ __launch_bounds__(1024)
void cosine_loss_kernel(const float* __restrict__ hfr, float* __restrict__ out)
{
    __shared__ float norms[BTDIM];
    __shared__ float red[1024];
    const int tid  = threadIdx.x;
    const int lane = tid & 31;
    const int wv   = tid >> 5;        // one wave32 per bt row

    float ss = 0.f;
    for (int c = lane; c < CCH; c += 32) {
        float v = hfr[wv * CCH + c];
        ss += v * v;
    }
    #pragma unroll
    for (int off = 16; off > 0; off >>= 1)
        ss += __shfl_xor(ss, off, 32);
    if (lane == 0) norms[wv] = __fsqrt_rn(ss);
    __syncthreads();

    float val = 0.f;
    if (tid < CCH) {
        float s = 0.f;
        #pragma unroll
        for (int b = 0; b < BTDIM; ++b)
            s += hfr[b * CCH + tid] / fmaxf(norms[b], 1e-12f);
        val = s * s;
    }
    red[tid] = val;
    __syncthreads();
    for (int s = 512; s > 0; s >>= 1) {
        if (tid < s) red[tid] += red[tid + s];
        __syncthreads();
    }
    if (tid == 0) out[0] = 1.0f - red[0] * (1.0f / 1024.0f);
}

// ---------------------------------------------------------------------------
extern "C" void kernel_launch(void* const* d_in, const int* in_sizes, int n_in,
                              void* d_out, int out_size, void* d_ws, size_t ws_size,
                              hipStream_t stream) {
    const float* x  = (const float*)d_in[0];   // (8, 256, 65536) fp32
    float* out = (float*)d_out;                // scalar loss
    float* hfr = (float*)d_ws;                 // 8192 floats of scratch

    spectral_band_kernel<<<dim3(BTDIM * CCH), dim3(128), 0, stream>>>(x, hfr);
    cosine_loss_kernel<<<dim3(1), dim3(1024), 0, stream>>>(hfr, out);
}